// SSMKernel_30365418783279
// MI455X (gfx1250) — compile-verified
//
#include <hip/hip_runtime.h>
#include <hip/hip_bf16.h>

#define NH 256
#define NN 64
#define LL 2048
#define CHUNK 16
#define NCHUNK (LL / CHUNK)   // 128

typedef __attribute__((ext_vector_type(2))) float v2f;
typedef __attribute__((ext_vector_type(8))) float v8f;

struct cpx { float r, i; };
__device__ __forceinline__ cpx cmul(cpx a, cpx b) { return {a.r*b.r - a.i*b.i, a.r*b.i + a.i*b.r}; }
__device__ __forceinline__ cpx cadd(cpx a, cpx b) { return {a.r + b.r, a.i + b.i}; }
__device__ __forceinline__ cpx csub(cpx a, cpx b) { return {a.r - b.r, a.i - b.i}; }
__device__ __forceinline__ cpx cinv(cpx a) { float d = a.r*a.r + a.i*a.i; return {a.r/d, -a.i/d}; }
__device__ __forceinline__ cpx cscale(cpx a, float s) { return {a.r*s, a.i*s}; }

// One 16x16 complex output tile of C = A(64xK cplx) @ B(Kx16/64 cplx),
// as 4 real f32 WMMA accumulation chains over K in steps of 4.
// Must be called with all 32 lanes of the wave active (EXEC all ones).
__device__ __forceinline__ void cplx_mm_tile(
    const float* Ar, const float* Ai, int lda,
    const float* Br, const float* Bi, int ldb,
    float* Cr, float* Ci, int ldc,
    int r0, int n0, int K)
{
    const int lane = threadIdx.x & 31;
    const int mn = lane & 15;            // row for A, col for B/C/D
    const int kh = (lane >> 4) << 1;     // k sub-offset: 0 or 2
    v8f arbr = {}; v8f aibi = {}; v8f arbi = {}; v8f aibr = {};
    for (int k0 = 0; k0 < K; k0 += 4) {
        const int ka = k0 + kh;
        v2f ar, ai, br, bi;
        ar[0] = Ar[(r0 + mn) * lda + ka];     ar[1] = Ar[(r0 + mn) * lda + ka + 1];
        ai[0] = Ai[(r0 + mn) * lda + ka];     ai[1] = Ai[(r0 + mn) * lda + ka + 1];
        br[0] = Br[ka * ldb + n0 + mn];       br[1] = Br[(ka + 1) * ldb + n0 + mn];
        bi[0] = Bi[ka * ldb + n0 + mn];       bi[1] = Bi[(ka + 1) * ldb + n0 + mn];
        arbr = __builtin_amdgcn_wmma_f32_16x16x4_f32(false, ar, false, br, (short)0, arbr, false, false);
        aibi = __builtin_amdgcn_wmma_f32_16x16x4_f32(false, ai, false, bi, (short)0, aibi, false, false);
        arbi = __builtin_amdgcn_wmma_f32_16x16x4_f32(false, ar, false, bi, (short)0, arbi, false, false);
        aibr = __builtin_amdgcn_wmma_f32_16x16x4_f32(false, ai, false, br, (short)0, aibr, false, false);
    }
    const int mbase = (lane >> 4) << 3;  // 0 or 8
    for (int v = 0; v < 8; ++v) {
        Cr[(r0 + mbase + v) * ldc + n0 + mn] = arbr[v] - aibi[v];
        Ci[(r0 + mbase + v) * ldc + n0 + mn] = arbi[v] + aibr[v];
    }
}

extern "C" __global__ __launch_bounds__(128)
void s4_dplr_kernel(const float* __restrict__ A_real, const float* __restrict__ A_imag,
                    const float* __restrict__ B_real, const float* __restrict__ B_imag,
                    const float* __restrict__ P_real, const float* __restrict__ P_imag,
                    const float* __restrict__ C_real, const float* __restrict__ C_imag,
                    const float* __restrict__ log_dt, float* __restrict__ out)
{
    extern __shared__ float sm[];
    const int h   = blockIdx.x;
    const int tid = threadIdx.x;
    const int wv  = tid >> 5;

    // LDS layout (floats)
    float* A0r = sm;                    // 4096
    float* A0i = sm + 4096;             // 4096
    float* A1r = sm + 8192;             // 4096
    float* A1i = sm + 12288;            // 4096
    float* Xr0 = sm + 16384;            // 1024
    float* Xi0 = sm + 17408;            // 1024
    float* Xr1 = sm + 18432;            // 1024
    float* Xi1 = sm + 19456;            // 1024
    float* ur  = sm + 20480;  float* ui = ur + 64;
    float* wr  = ui + 64;     float* wi = wr + 64;
    float* gr  = wi + 64;     float* gi = gr + 64;
    float* cr  = gi + 64;     float* ci = cr + 64;
    float* pr  = ci + 64;     float* pi = pr + 64;
    float* scal = pi + 64;              // 8 scalars

    const float dt = __expf(log_dt[h]);
    const float cc = 0.5f * dt;

    // ---------- Woodbury setup:  M = D + c p p^H,  D = diag(1 - c*lam) ----------
    if (tid < 64) {
        const int n = tid;
        cpx lam = { -A_real[h*64 + n], -A_imag[h*64 + n] };
        cpx p   = {  P_real[h*64 + n],  P_imag[h*64 + n] };
        cpx Dinv = cinv((cpx){1.0f - cc*lam.r, -cc*lam.i});
        cpx e    = {1.0f + cc*lam.r,  cc*lam.i};
        cpx u = cmul(Dinv, p);
        cpx g = cmul(Dinv, e);
        ur[n] = u.r; ui[n] = u.i; gr[n] = g.r; gi[n] = g.i;
        pr[n] = p.r; pi[n] = p.i;
        cr[n] = C_real[h*64 + n]; ci[n] = C_imag[h*64 + n];
    }
    __syncthreads();
    if (tid == 0) {
        cpx s = {0.f, 0.f};
        for (int n = 0; n < 64; ++n)
            s = cadd(s, cmul((cpx){pr[n], -pi[n]}, (cpx){ur[n], ui[n]}));   // s = p^H D^-1 p
        cpx alpha = cmul((cpx){cc, 0.f}, cinv((cpx){1.0f + cc*s.r, cc*s.i}));
        cpx base  = csub((cpx){cc, 0.f}, cmul(alpha, cscale(s, cc)));       // c - alpha*c*s
        scal[0] = alpha.r; scal[1] = alpha.i; scal[2] = base.r; scal[3] = base.i;
    }
    __syncthreads();
    const cpx alpha = {scal[0], scal[1]};
    const cpx base  = {scal[2], scal[3]};
    if (tid < 64) {
        const int n = tid;
        // w_m = conj(p_m) * (base + alpha*g_m)  ->  dA = diag(g) - u w^T
        cpx w = cmul((cpx){pr[n], -pi[n]}, cadd(base, cmul(alpha, (cpx){gr[n], gi[n]})));
        wr[n] = w.r; wi[n] = w.i;
    }
    __syncthreads();
    for (int idx = tid; idx < 4096; idx += 128) {
        const int n = idx >> 6, m = idx & 63;
        cpx v = cmul((cpx){ur[n], ui[n]}, (cpx){wr[m], wi[m]});
        float rr = -v.r, rim = -v.i;
        if (n == m) { rr += gr[n]; rim += gi[n]; }
        A0r[idx] = rr; A0i[idx] = rim;
    }
    __syncthreads();

    // ---------- dB = M^{-1}(dt*B) via same Sherman–Morrison ----------
    if (tid < 64) {
        const int n = tid;
        cpx lam = { -A_real[h*64 + n], -A_imag[h*64 + n] };
        cpx Dinv = cinv((cpx){1.0f - cc*lam.r, -cc*lam.i});
        cpx b = { B_real[h*64 + n], B_imag[h*64 + n] };
        cpx t = cmul(Dinv, cscale(b, dt));
        wr[n] = t.r; wi[n] = t.i;   // w no longer needed: reuse as t
    }
    __syncthreads();
    if (tid == 0) {
        cpx q = {0.f, 0.f};
        for (int n = 0; n < 64; ++n)
            q = cadd(q, cmul((cpx){pr[n], -pi[n]}, (cpx){wr[n], wi[n]}));
        cpx aq = cmul(alpha, q);
        scal[4] = aq.r; scal[5] = aq.i;
    }
    __syncthreads();
    if (tid < 64) {
        const int n = tid;
        cpx dB = csub((cpx){wr[n], wi[n]}, cmul((cpx){scal[4], scal[5]}, (cpx){ur[n], ui[n]}));
        Xr0[n*16 + 0] = dB.r;
        Xi0[n*16 + 0] = dB.i;
    }
    __syncthreads();

    // ---------- first 16 Krylov columns: x_l = dA x_{l-1} (VALU matvecs) ----------
    for (int l = 1; l < CHUNK; ++l) {
        if (tid < 64) {
            const int n = tid;
            cpx acc = {0.f, 0.f};
            for (int m = 0; m < 64; ++m)
                acc = cadd(acc, cmul((cpx){A0r[n*64 + m], A0i[n*64 + m]},
                                     (cpx){Xr0[m*16 + l - 1], Xi0[m*16 + l - 1]}));
            Xr0[n*16 + l] = acc.r; Xi0[n*16 + l] = acc.i;
        }
        __syncthreads();
    }
    if (tid < 16) {
        float k = 0.f;
        for (int n = 0; n < 64; ++n)
            k += cr[n]*Xr0[n*16 + tid] - ci[n]*Xi0[n*16 + tid];
        out[h*LL + tid] = 2.0f * k;
    }
    __syncthreads();

    // ---------- dA^16 by 4 WMMA squarings, fully explicit (no LDS pointer phis) ----------
    // A0 -> A1 (dA^2), A1 -> A0 (dA^4), A0 -> A1 (dA^8), A1 -> A0 (dA^16)
    {
        for (int n0 = 0; n0 < 64; n0 += 16)
            cplx_mm_tile(A0r, A0i, 64, A0r, A0i, 64, A1r, A1i, 64, 16*wv, n0, 64);
        __syncthreads();
        for (int n0 = 0; n0 < 64; n0 += 16)
            cplx_mm_tile(A1r, A1i, 64, A1r, A1i, 64, A0r, A0i, 64, 16*wv, n0, 64);
        __syncthreads();
        for (int n0 = 0; n0 < 64; n0 += 16)
            cplx_mm_tile(A0r, A0i, 64, A0r, A0i, 64, A1r, A1i, 64, 16*wv, n0, 64);
        __syncthreads();
        for (int n0 = 0; n0 < 64; n0 += 16)
            cplx_mm_tile(A1r, A1i, 64, A1r, A1i, 64, A0r, A0i, 64, 16*wv, n0, 64);
        __syncthreads();
    }
    // dA^16 now in A0r/A0i

    // ---------- main scan: X_j = dA^16 @ X_{j-1}, K[16j..16j+15] = 2 Re(c^T X_j) ----------
    // Double-stepped so every cplx_mm_tile call site has compile-time LDS pointers.
    auto proj_out = [&](const float* Dxr, const float* Dxi, int j) {
        if (tid < 16) {
            float k = 0.f;
            for (int n = 0; n < 64; ++n)
                k += cr[n]*Dxr[n*16 + tid] - ci[n]*Dxi[n*16 + tid];
            out[h*LL + j*CHUNK + tid] = 2.0f * k;
        }
        __syncthreads();
    };

    for (int jj = 0; jj < 63; ++jj) {
        const int j = 2*jj + 1;
        // X0 -> X1
        cplx_mm_tile(A0r, A0i, 64, Xr0, Xi0, 16, Xr1, Xi1, 16, 16*wv, 0, 64);
        __syncthreads();
        proj_out(Xr1, Xi1, j);
        // X1 -> X0
        cplx_mm_tile(A0r, A0i, 64, Xr1, Xi1, 16, Xr0, Xi0, 16, 16*wv, 0, 64);
        __syncthreads();
        proj_out(Xr0, Xi0, j + 1);
    }
    // final step: j = 127, X0 -> X1
    cplx_mm_tile(A0r, A0i, 64, Xr0, Xi0, 16, Xr1, Xi1, 16, 16*wv, 0, 64);
    __syncthreads();
    proj_out(Xr1, Xi1, 127);
}

extern "C" void kernel_launch(void* const* d_in, const int* in_sizes, int n_in,
                              void* d_out, int out_size, void* d_ws, size_t ws_size,
                              hipStream_t stream) {
    const float* A_real = (const float*)d_in[0];
    const float* A_imag = (const float*)d_in[1];
    const float* B_real = (const float*)d_in[2];
    const float* B_imag = (const float*)d_in[3];
    const float* P_real = (const float*)d_in[4];
    const float* P_imag = (const float*)d_in[5];
    const float* C_real = (const float*)d_in[6];
    const float* C_imag = (const float*)d_in[7];
    const float* log_dt = (const float*)d_in[8];
    float* out = (float*)d_out;
    (void)in_sizes; (void)n_in; (void)out_size; (void)d_ws; (void)ws_size;

    const size_t shmem_bytes = 21128u * sizeof(float);  // ~84.5 KB (WGP has 320 KB LDS)
    s4_dplr_kernel<<<NH, 128, shmem_bytes, stream>>>(
        A_real, A_imag, B_real, B_imag, P_real, P_imag, C_real, C_imag, log_dt, out);
}